// Model_32177894982138
// MI455X (gfx1250) — compile-verified
//
#include <hip/hip_runtime.h>

#define NNODES 50000
#define NEDGES 250000
#define DIM    300
#define NGRAPH 1000
#define NHOPS  4

#define CDIV(a,b) (((a)+(b)-1)/(b))

typedef __attribute__((ext_vector_type(2))) float v2f;
typedef __attribute__((ext_vector_type(8))) float v8f;

// ---------------- elementwise / graph-prep kernels ----------------

__global__ void fill_kernel(float* __restrict__ p, float v, int n) {
    int i = blockIdx.x * blockDim.x + threadIdx.x;
    if (i < n) p[i] = v;
}

__global__ void embed_kernel(const int* __restrict__ x,
                             const float* __restrict__ at,
                             const float* __restrict__ wt,
                             float* __restrict__ h) {
    int i = blockIdx.x * blockDim.x + threadIdx.x;
    if (i >= NNODES * DIM) return;
    int n = i / DIM, d = i - n * DIM;
    int a = x[2 * n + 0];
    int w = x[2 * n + 1];
    h[i] = at[(size_t)a * DIM + d] + wt[(size_t)w * DIM + d];
}

__global__ void deg_kernel(const int* __restrict__ dst, float* __restrict__ deg) {
    int e = blockIdx.x * blockDim.x + threadIdx.x;
    if (e < NEDGES) atomicAdd(&deg[dst[e]], 1.0f);
}

__global__ void dinv_kernel(float* __restrict__ deg) {
    int n = blockIdx.x * blockDim.x + threadIdx.x;
    if (n < NNODES) deg[n] = rsqrtf(deg[n]);   // deg >= 1 (self loop) always
}

__global__ void norm_kernel(const int* __restrict__ src, const int* __restrict__ dst,
                            const float* __restrict__ dinv, float* __restrict__ norm) {
    int e = blockIdx.x * blockDim.x + threadIdx.x;
    if (e < NEDGES) norm[e] = dinv[src[e]] * dinv[dst[e]];
}

// h[n,:] = b[:] + dinv[n]^2 * hl[n,:]   (bias + self-loop term; overwrites h in place)
__global__ void selfinit_kernel(const float* __restrict__ hl, const float* __restrict__ dinv,
                                const float* __restrict__ b, float* __restrict__ h) {
    int i = blockIdx.x * blockDim.x + threadIdx.x;
    if (i >= NNODES * DIM) return;
    int n = i / DIM, d = i - n * DIM;
    float s = dinv[n];
    h[i] = b[d] + s * s * hl[i];
}

// one thread per (edge, 4-float chunk): h[dst,:] += norm[e] * hl[src,:]
__global__ void scatter_kernel(const float* __restrict__ hl, const int* __restrict__ src,
                               const int* __restrict__ dst, const float* __restrict__ norm,
                               float* __restrict__ h) {
    int t = blockIdx.x * blockDim.x + threadIdx.x;
    if (t >= NEDGES * (DIM / 4)) return;
    int e = t / (DIM / 4);
    int c = (t - e * (DIM / 4)) * 4;
    float w = norm[e];
    const float4 v = *(const float4*)(hl + (size_t)src[e] * DIM + c);
    float* o = h + (size_t)dst[e] * DIM + c;
    atomicAdd(o + 0, w * v.x);
    atomicAdd(o + 1, w * v.y);
    atomicAdd(o + 2, w * v.z);
    atomicAdd(o + 3, w * v.w);
}

__global__ void pool_sum_kernel(const float* __restrict__ h, const int* __restrict__ batch,
                                float* __restrict__ out) {
    int i = blockIdx.x * blockDim.x + threadIdx.x;
    if (i >= NNODES * DIM) return;
    int n = i / DIM, d = i - n * DIM;
    atomicAdd(&out[(size_t)batch[n] * DIM + d], h[i]);
}

__global__ void pool_cnt_kernel(const int* __restrict__ batch, float* __restrict__ counts) {
    int n = blockIdx.x * blockDim.x + threadIdx.x;
    if (n < NNODES) atomicAdd(&counts[batch[n]], 1.0f);
}

__global__ void pool_div_kernel(float* __restrict__ out, const float* __restrict__ counts) {
    int i = blockIdx.x * blockDim.x + threadIdx.x;
    if (i >= NGRAPH * DIM) return;
    int g = i / DIM;
    float c = counts[g];
    out[i] = out[i] / (c < 1.0f ? 1.0f : c);
}

// ---------------- WMMA fp32 GEMM: Cout[M,300] = A[M,300] @ W[300,300] ----------------
// Block: 256 threads = 8 waves; each wave owns a 16-row M strip (128 rows per block).
// W is staged in LDS *pre-swizzled into WMMA B-fragment layout*:
//   lw[(g*304 + c)*4 + q] = W[kk + 4g + q][c]      (g = k-group of 4, q = k within group)
// so a lane's B fragment is ONE aligned ds_load_b64 at float index (g*304+col)*4 + 2*hi;
// all g/t deltas fold into DS immediate offsets; DS reads hit all 64 banks exactly once.
// B-fragment loads are software-pipelined one pair ahead so DS latency hides behind WMMAs.
// K padded to 320 (10 chunks of 32, zeros staged beyond k=299) for branch-free unrolling.

#define KC    32
#define KG    8     // k-groups of 4 per chunk
#define LDSW  304
#define NT    19    // ceil(300/16)

__global__ __launch_bounds__(256)
void gemm_wmma_kernel(const float* __restrict__ A, const float* __restrict__ Wm,
                      float* __restrict__ Cout, int M) {
    __shared__ float lw[KC * LDSW];              // 38912 bytes

    const int lane = threadIdx.x & 31;
    const int wave = threadIdx.x >> 5;
    const int hi   = lane >> 4;                  // 0: lanes 0-15, 1: lanes 16-31
    const int lmod = lane & 15;
    const int m0   = blockIdx.x * 128 + wave * 16;
    int mrow = m0 + lmod;
    if (mrow > M - 1) mrow = M - 1;              // clamp: garbage rows are never stored
    const float* arow = A + (size_t)mrow * DIM;

    // per-lane LDS base (floats): fragment origin for g=0, t=0
    const int lbase = lmod * 4 + hi * 2;

    v8f acc[NT];
    #pragma unroll
    for (int t = 0; t < NT; ++t) {
        #pragma unroll
        for (int j = 0; j < 8; ++j) acc[t][j] = 0.0f;
    }

    for (int kk = 0; kk < 320; kk += KC) {       // 10 chunks; last 20 K-slices are zeros
        __syncthreads();
        // cooperative stage: coalesced global reads, swizzled LDS writes
        for (int i = threadIdx.x; i < KC * LDSW; i += 256) {
            int kl = i / LDSW;                   // 0..31
            int c  = i - kl * LDSW;              // 0..303
            int g  = kl >> 2;
            int q  = kl & 3;
            int k  = kk + kl;
            lw[(g * LDSW + c) * 4 + q] =
                (k < DIM && c < DIM) ? Wm[(size_t)k * DIM + c] : 0.0f;
        }
        __syncthreads();

        #pragma unroll
        for (int g = 0; g < KG; ++g) {
            const int kb = kk + 4 * g + 2 * hi;  // A K index (reads past 300 hit zeros in B)
            const v2f av = *(const v2f*)(arow + kb);
            const float* lbp = &lw[g * (LDSW * 4) + lbase];

            // software-pipelined B-fragment stream: keep the next pair in flight
            v2f b0 = *(const v2f*)(lbp + 0 * 64);
            v2f b1 = *(const v2f*)(lbp + 1 * 64);
            #pragma unroll
            for (int t = 0; t < NT; t += 2) {
                const v2f c0 = b0;
                const v2f c1 = b1;
                if (t + 2 < NT) b0 = *(const v2f*)(lbp + (t + 2) * 64);
                if (t + 3 < NT) b1 = *(const v2f*)(lbp + (t + 3) * 64);
                acc[t] = __builtin_amdgcn_wmma_f32_16x16x4_f32(
                    false, av, false, c0, (short)0, acc[t], false, false);
                if (t + 1 < NT) {
                    acc[t + 1] = __builtin_amdgcn_wmma_f32_16x16x4_f32(
                        false, av, false, c1, (short)0, acc[t + 1], false, false);
                }
            }
        }
    }

    // store: C/D layout — VGPR j: lanes 0-15 => row m0+j, lanes 16-31 => row m0+j+8
    #pragma unroll
    for (int t = 0; t < NT; ++t) {
        const int col = t * 16 + lmod;
        if (col < DIM) {
            #pragma unroll
            for (int j = 0; j < 8; ++j) {
                const int row = m0 + j + 8 * hi;
                if (row < M) Cout[(size_t)row * DIM + col] = acc[t][j];
            }
        }
    }
}

// ---------------- launch ----------------

extern "C" void kernel_launch(void* const* d_in, const int* in_sizes, int n_in,
                              void* d_out, int out_size, void* d_ws, size_t ws_size,
                              hipStream_t stream) {
    const int*   x     = (const int*)d_in[0];     // [N,2]
    const int*   ei    = (const int*)d_in[1];     // [2,E]
    const int*   batch = (const int*)d_in[2];     // [N]
    // d_in[3]: num_hops (device scalar; fixed at 4 by setup -> hardcoded for capture)
    const float* atomT = (const float*)d_in[4];   // [5001,300]
    const float* wordT = (const float*)d_in[5];   // [30000,300]
    const float* Wm    = (const float*)d_in[6];   // [300,300]
    const float* bias  = (const float*)d_in[7];   // [300]
    float*       out   = (float*)d_out;           // [1000,300]

    const int* src = ei;
    const int* dst = ei + NEDGES;

    float* ws     = (float*)d_ws;
    float* h      = ws;                               // N*D
    float* hl     = h  + (size_t)NNODES * DIM;        // N*D
    float* dinv   = hl + (size_t)NNODES * DIM;        // N (deg, then rsqrt in place)
    float* norm   = dinv + NNODES;                    // E
    float* counts = norm + NEDGES;                    // G

    const int B = 256;

    // node embeddings
    embed_kernel<<<CDIV(NNODES * DIM, B), B, 0, stream>>>(x, atomT, wordT, h);

    // degrees (init 1.0 for self loop), dinv, edge norms
    fill_kernel<<<CDIV(NNODES, B), B, 0, stream>>>(dinv, 1.0f, NNODES);
    deg_kernel<<<CDIV(NEDGES, B), B, 0, stream>>>(dst, dinv);
    dinv_kernel<<<CDIV(NNODES, B), B, 0, stream>>>(dinv);
    norm_kernel<<<CDIV(NEDGES, B), B, 0, stream>>>(src, dst, dinv, norm);

    // 4 GCN hops
    for (int hop = 0; hop < NHOPS; ++hop) {
        gemm_wmma_kernel<<<CDIV(NNODES, 128), B, 0, stream>>>(h, Wm, hl, NNODES);
        selfinit_kernel<<<CDIV(NNODES * DIM, B), B, 0, stream>>>(hl, dinv, bias, h);
        scatter_kernel<<<CDIV(NEDGES * (DIM / 4), B), B, 0, stream>>>(hl, src, dst, norm, h);
    }

    // global mean pool
    fill_kernel<<<CDIV(NGRAPH * DIM, B), B, 0, stream>>>(out, 0.0f, NGRAPH * DIM);
    fill_kernel<<<CDIV(NGRAPH, B), B, 0, stream>>>(counts, 0.0f, NGRAPH);
    pool_sum_kernel<<<CDIV(NNODES * DIM, B), B, 0, stream>>>(h, batch, out);
    pool_cnt_kernel<<<CDIV(NNODES, B), B, 0, stream>>>(batch, counts);
    pool_div_kernel<<<CDIV(NGRAPH * DIM, B), B, 0, stream>>>(out, counts);
}